// GCNNet_43404939493621
// MI455X (gfx1250) — compile-verified
//
#include <hip/hip_runtime.h>
#include <cstdint>
#include <cstddef>

typedef float v2f __attribute__((ext_vector_type(2)));
typedef float v8f __attribute__((ext_vector_type(8)));

// Native fire-and-forget f32 atomic add (global_atomic_add_f32).
__device__ __forceinline__ void atomic_add_f32(float* p, float v) {
  unsafeAtomicAdd(p, v);
}

// ---------------------------------------------------------------------------
// Degree / normalization kernels
// ---------------------------------------------------------------------------
__global__ void fill_f32(float* __restrict__ p, float v, int n) {
  int i = blockIdx.x * blockDim.x + threadIdx.x;
  if (i < n) p[i] = v;
}

__global__ void deg_accum(const int* __restrict__ dst, float* __restrict__ deg, int E) {
  int i = blockIdx.x * blockDim.x + threadIdx.x;
  if (i < E) atomic_add_f32(&deg[dst[i]], 1.0f);
}

__global__ void rsqrt_inplace(float* __restrict__ p, int n) {
  int i = blockIdx.x * blockDim.x + threadIdx.x;
  if (i < n) {
    float d = p[i];
    p[i] = (d > 0.0f) ? (1.0f / sqrtf(d)) : 0.0f;
  }
}

// ---------------------------------------------------------------------------
// WMMA GEMM with LDS-staged B: C[M,N] = (A[M,K] @ B[K,N]) * dinv[row]
// fp32 V_WMMA_F32_16X16X4_F32; wave32; TN = N/16 output tiles per wave.
// Each wave owns one 16-row M-tile and accumulates ALL N-tiles, so each A
// element is loaded from global exactly once (as an aligned float2/b64) and
// all B fragments are served from LDS (whole W staged once per block).
//
// Fragment layouts per CDNA5 ISA 7.12.2 (wave32):
//   A 16x4 : lanes 0-15 -> M=lane, v0=K0 v1=K1 ; lanes 16-31 -> v0=K2 v1=K3
//   B 4x16 : lanes 0-15 -> N=lane, v0=K0 v1=K1 ; lanes 16-31 -> v0=K2 v1=K3
//   C 16x16: VGPR r, lane l -> row = r + 8*(l>>4), col = l&15
// ---------------------------------------------------------------------------
template <int TN>
__global__ void gemm_wmma_lds(const float* __restrict__ A,
                              const float* __restrict__ Bw,
                              const float* __restrict__ dinv,
                              float* __restrict__ C,
                              int M, int K) {
  const int N = TN * 16;
  extern __shared__ float ldsW[];  // K*N floats (<= 32KB here)

  // Cooperative stage of W into LDS (vectorized float4 copy).
  {
    const int totalV4 = (K * N) >> 2;
    const float4* __restrict__ src4 = (const float4*)Bw;
    float4* dst4 = (float4*)ldsW;
    for (int i = threadIdx.x; i < totalV4; i += blockDim.x) dst4[i] = src4[i];
  }
  __syncthreads();

  const int lane = threadIdx.x & 31;
  const int half = lane >> 4;   // 0: K0/K1 side, 1: K2/K3 side
  const int l16  = lane & 15;
  const int wavesPerBlock = blockDim.x >> 5;
  const int wave   = blockIdx.x * wavesPerBlock + (threadIdx.x >> 5);
  const int nWaves = gridDim.x * wavesPerBlock;
  const int tilesM = M >> 4;

  for (int tm = wave; tm < tilesM; tm += nWaves) {
    const int row0 = tm << 4;
    int arow_i = row0 + l16;
    if (arow_i >= M) arow_i = M - 1;   // ragged-M clamp (read-only safety)
    const float* __restrict__ arow = A + (size_t)arow_i * K;

    v8f acc[TN];
#pragma unroll
    for (int tn = 0; tn < TN; ++tn) acc[tn] = (v8f){0.f,0.f,0.f,0.f,0.f,0.f,0.f,0.f};

    for (int k0 = 0; k0 < K; k0 += 4) {
      const int ka = k0 + (half << 1);
      // A fragment: one aligned 8-byte load (K is a multiple of 4, ka even)
      const float2 a2 = *(const float2*)(arow + ka);
      v2f afrag; afrag.x = a2.x; afrag.y = a2.y;
      const float* wrow0 = ldsW + ka * N;
#pragma unroll
      for (int tn = 0; tn < TN; ++tn) {
        v2f bfrag;
        bfrag.x = wrow0[tn * 16 + l16];
        bfrag.y = wrow0[N + tn * 16 + l16];
        acc[tn] = __builtin_amdgcn_wmma_f32_16x16x4_f32(
            false, afrag, false, bfrag, (short)0, acc[tn], false, false);
      }
    }

#pragma unroll
    for (int r = 0; r < 8; ++r) {
      const int row = row0 + r + (half << 3);
      if (row < M) {
        const float s = dinv[row];
        float* crow = C + (size_t)row * N;
#pragma unroll
        for (int tn = 0; tn < TN; ++tn) crow[tn * 16 + l16] = acc[tn][r] * s;
      }
    }
  }
}

// Generic fallback (one 16x16 tile per wave), used only for shapes where
// N/16 is not 4 or 8, or where W (K*N*4 bytes) exceeds 160KB of LDS.
__global__ void gemm_wmma_generic(const float* __restrict__ A,
                                  const float* __restrict__ Bw,
                                  const float* __restrict__ dinv,
                                  float* __restrict__ C,
                                  int M, int K, int N) {
  const int lane = threadIdx.x & 31;
  const int half = lane >> 4;
  const int l16  = lane & 15;
  const int wavesPerBlock = blockDim.x >> 5;
  const int wave   = blockIdx.x * wavesPerBlock + (threadIdx.x >> 5);
  const int nWaves = gridDim.x * wavesPerBlock;
  const int tilesN = N >> 4;
  const int totalTiles = (M >> 4) * tilesN;

  for (int t = wave; t < totalTiles; t += nWaves) {
    const int tm = t / tilesN;
    const int tn = t - tm * tilesN;
    const int row0 = tm << 4, col0 = tn << 4;
    int arow_i = row0 + l16;
    if (arow_i >= M) arow_i = M - 1;
    const float* __restrict__ arow = A + (size_t)arow_i * K;
    v8f acc = {0.f,0.f,0.f,0.f,0.f,0.f,0.f,0.f};
    for (int k0 = 0; k0 < K; k0 += 4) {
      const int ka = k0 + (half << 1);
      const float2 a2 = *(const float2*)(arow + ka);
      v2f afrag; afrag.x = a2.x; afrag.y = a2.y;
      v2f bfrag;
      bfrag.x = Bw[(size_t)ka * N + col0 + l16];
      bfrag.y = Bw[(size_t)(ka + 1) * N + col0 + l16];
      acc = __builtin_amdgcn_wmma_f32_16x16x4_f32(
          false, afrag, false, bfrag, (short)0, acc, false, false);
    }
#pragma unroll
    for (int r = 0; r < 8; ++r) {
      const int row = row0 + r + (half << 3);
      if (row < M) C[(size_t)row * N + col0 + l16] = acc[r] * dinv[row];
    }
  }
}

// ---------------------------------------------------------------------------
// Edge aggregation: acc[dst] += hs[src] (hs already pre-scaled by dinv[src]).
// Self-loops appended logically as edges [E, E+N). One thread per float4.
// The edge-index stream for the NEXT grid-stride iteration is prefetched
// (global_prefetch_b8) to overlap the two-hop index->gather dependency with
// the current iteration's atomic traffic.
// ---------------------------------------------------------------------------
__global__ void gather_scatter(const float* __restrict__ hs,
                               const int* __restrict__ src,
                               const int* __restrict__ dst,
                               float* __restrict__ acc,
                               int E, int N, int F, int qshift) {
  const int quads = F >> 2;
  const long long total  = (long long)(E + N) << qshift;
  const long long stride = (long long)gridDim.x * blockDim.x;
  for (long long idx = (long long)blockIdx.x * blockDim.x + threadIdx.x;
       idx < total; idx += stride) {
    // Prefetch edge indices for the next grid-stride iteration.
    const long long nidx = idx + stride;
    if (nidx < total) {
      const int ne = (int)(nidx >> qshift);
      if (ne < E) {
        __builtin_prefetch(&src[ne], 0, 0);
        __builtin_prefetch(&dst[ne], 0, 0);
      }
    }

    const int e = (int)(idx >> qshift);
    const int q = (int)(idx & (quads - 1));
    int s, d;
    if (e < E) { s = src[e]; d = dst[e]; }
    else       { s = e - E;  d = s; }
    const float4 v = *(const float4*)(hs + (size_t)s * F + (q << 2));
    float* p = acc + (size_t)d * F + (q << 2);
    atomic_add_f32(p + 0, v.x);
    atomic_add_f32(p + 1, v.y);
    atomic_add_f32(p + 2, v.z);
    atomic_add_f32(p + 3, v.w);
  }
}

// ---------------------------------------------------------------------------
// out[i,j] = relu(dinv[i] * acc[i,j] + bias[j]); F power of two.
// NOTE: `out` may alias `acc` (layer-1 runs in place), so neither carries
// __restrict__.
// ---------------------------------------------------------------------------
__global__ void bias_relu_scale(const float* acc,
                                const float* __restrict__ dinv,
                                const float* __restrict__ bias,
                                float* out,
                                int N, int F, int fshift) {
  const long long total  = (long long)N << fshift;
  const long long stride = (long long)gridDim.x * blockDim.x;
  for (long long idx = (long long)blockIdx.x * blockDim.x + threadIdx.x;
       idx < total; idx += stride) {
    const int i = (int)(idx >> fshift);
    const int j = (int)(idx & (F - 1));
    const float v = fmaf(dinv[i], acc[idx], bias[j]);
    out[idx] = fmaxf(v, 0.0f);
  }
}

// ---------------------------------------------------------------------------
static inline int ilog2(int x) { return 31 - __builtin_clz((unsigned)x); }

static void launch_gemm(const float* A, const float* W, const float* dinv,
                        float* C, int M, int K, int N, hipStream_t stream) {
  const int T = 256;
  const int tilesM = M >> 4;
  const int tilesN = N >> 4;
  const int wavesPerBlock = T >> 5;
  const size_t shmem = (size_t)K * N * sizeof(float);
  const int blocks = (tilesM + wavesPerBlock - 1) / wavesPerBlock;
  if (tilesN == 8 && shmem <= 160 * 1024) {
    gemm_wmma_lds<8><<<blocks, T, shmem, stream>>>(A, W, dinv, C, M, K);
  } else if (tilesN == 4 && shmem <= 160 * 1024) {
    gemm_wmma_lds<4><<<blocks, T, shmem, stream>>>(A, W, dinv, C, M, K);
  } else {
    const int tiles = tilesM * tilesN;
    const int gblocks = (tiles + wavesPerBlock - 1) / wavesPerBlock;
    gemm_wmma_generic<<<gblocks, T, 0, stream>>>(A, W, dinv, C, M, K, N);
  }
}

extern "C" void kernel_launch(void* const* d_in, const int* in_sizes, int n_in,
                              void* d_out, int out_size, void* d_ws, size_t ws_size,
                              hipStream_t stream) {
  const float* x  = (const float*)d_in[0];
  const int*   ei = (const int*)d_in[1];
  const float* W1 = (const float*)d_in[2];
  const float* b1 = (const float*)d_in[3];
  const float* W2 = (const float*)d_in[4];
  const float* b2 = (const float*)d_in[5];
  float* out = (float*)d_out;

  const int hidden = in_sizes[3];            // 128
  const int outF   = in_sizes[5];            // 64
  const int inF    = in_sizes[2] / hidden;   // 64
  const int N      = in_sizes[0] / inF;      // 50000
  const int E      = in_sizes[1] / 2;        // 800000
  const int* srcp = ei;
  const int* dstp = ei + E;

  // Workspace layout (fp32):
  //   dinv : N                      (degree, then dinv in place)
  //   bufA : N*hidden               (h_scaled L1; reused for h2_scaled L2)
  //   bufB : N*hidden               (acc L1; then h1 after bias/relu, in place)
  //   bufD : N*outF                 (acc L2)
  char* ws = (char*)d_ws;
  float* dinv = (float*)ws;
  size_t offA = (((size_t)N * sizeof(float)) + 255) & ~(size_t)255;
  float* bufA = (float*)(ws + offA);
  float* bufB = bufA + (size_t)N * hidden;
  float* bufD = bufB + (size_t)N * hidden;

  const int T = 256;

  // Zero the two atomic accumulators (graph-capturable memset nodes).
  hipMemsetAsync(bufB, 0, (size_t)N * hidden * sizeof(float), stream);
  hipMemsetAsync(bufD, 0, (size_t)N * outF * sizeof(float), stream);

  // --- degrees (self-loop => start at 1.0) and dinv = 1/sqrt(deg) ---
  fill_f32<<<(N + T - 1) / T, T, 0, stream>>>(dinv, 1.0f, N);
  deg_accum<<<(E + T - 1) / T, T, 0, stream>>>(dstp, dinv, E);
  rsqrt_inplace<<<(N + T - 1) / T, T, 0, stream>>>(dinv, N);

  // --- layer 1: h1 = relu(dinv * scatter(dinv*x@W1) + b1) ---
  launch_gemm(x, W1, dinv, bufA, N, inF, hidden, stream);
  {
    const int qshift = ilog2(hidden >> 2);
    const long long tot = (long long)(E + N) << qshift;
    gather_scatter<<<(int)((tot + T - 1) / T), T, 0, stream>>>(
        bufA, srcp, dstp, bufB, E, N, hidden, qshift);
    const long long ft = (long long)N * hidden;
    bias_relu_scale<<<(int)((ft + T - 1) / T), T, 0, stream>>>(
        bufB, dinv, b1, bufB, N, hidden, ilog2(hidden));
  }

  // --- layer 2: out = relu(dinv * scatter(dinv*h1@W2) + b2) ---
  launch_gemm(bufB, W2, dinv, bufA, N, hidden, outF, stream);
  {
    const int qshift = ilog2(outF >> 2);
    const long long tot = (long long)(E + N) << qshift;
    gather_scatter<<<(int)((tot + T - 1) / T), T, 0, stream>>>(
        bufA, srcp, dstp, bufD, E, N, outF, qshift);
    const long long ft = (long long)N * outF;
    bias_relu_scale<<<(int)((ft + T - 1) / T), T, 0, stream>>>(
        bufD, dinv, b2, out, N, outF, ilog2(outF));
  }
}